// OmicsEmbedder_83296595738828
// MI455X (gfx1250) — compile-verified
//
#include <hip/hip_runtime.h>

typedef __attribute__((ext_vector_type(2))) float    v2f;
typedef __attribute__((ext_vector_type(8))) float    v8f;
typedef __attribute__((ext_vector_type(4))) unsigned v4u;

#define N_CELLS_C        4096
#define NUM_HID_C        128
#define CELLS_PER_BLOCK  16
#define CHUNK            512       // nnz staged per iteration (== blockDim)
#define THREADS          512       // 16 waves: 8 h-tiles x 2-way split-K

// One block = 16 cells x 128 hidden, 2-way split-K.
// Wave (kg, ht): kg = wave>>3 picks K-half of each staged chunk, ht = wave&7
// picks the 16-wide h-tile. D(16x16) += A(16x4 scatter-of-values) x
// B(4x16 gathered embedding slices) on V_WMMA_F32_16X16X4_F32; the two K-half
// accumulators are summed deterministically through LDS before the store.
__global__ __launch_bounds__(THREADS) void omics_spmm_wmma(
    const float* __restrict__ xval,
    const int*   __restrict__ xrow,   // sorted ascending
    const int*   __restrict__ xcol,
    const int*   __restrict__ gidx,
    const float* __restrict__ embs,   // [30000,128] f32, L2-resident (15.4 MB)
    float*       __restrict__ out,    // [4096,128] f32
    int nnz)
{
    // 8 bytes per staged nnz: {val bits, (row-base+1)<<27 | (g*128)}
    __shared__ alignas(16) uint2 s_vm[CHUNK];
    __shared__ float s_red[8 * 16 * 16];        // split-K partials, 8 KB

    const int tid  = threadIdx.x;
    const int lane = tid & 31;
    const int wv   = tid >> 5;                  // 0..15
    const int kg   = wv >> 3;                   // split-K group 0/1
    const int ht   = wv & 7;                    // h-tile
    const int base = blockIdx.x * CELLS_PER_BLOCK;

    // Contiguous nnz range of this block's 16 cells (rows sorted). Result is
    // wave-uniform; readfirstlane makes that visible -> scalar loop control,
    // EXEC stays all-1s around the WMMAs (ISA requirement).
    int lo = 0, hi = nnz;
    while (lo < hi) { int mid = (lo + hi) >> 1; if (xrow[mid] < base) lo = mid + 1; else hi = mid; }
    const int segStart = __builtin_amdgcn_readfirstlane(lo);
    hi = nnz;
    while (lo < hi) { int mid = (lo + hi) >> 1; if (xrow[mid] < base + CELLS_PER_BLOCK) lo = mid + 1; else hi = mid; }
    const int segEnd = __builtin_amdgcn_readfirstlane(lo);

    v8f acc = {0.f,0.f,0.f,0.f,0.f,0.f,0.f,0.f};

    const unsigned m1   = (lane & 15) + 1;      // matches stored row+1
    const unsigned hcol = ht * 16 + (lane & 15);
    const int      ksel = (lane >> 4) << 1;     // lanes 0-15: K=0,1 ; 16-31: K=2,3
    const int      kbeg = __builtin_amdgcn_readfirstlane(kg * (CHUNK / 2));

    for (int ck = segStart; ck < segEnd; ck += CHUNK) {
        __syncthreads();                        // previous chunk fully consumed
        const int idx = ck + tid;
        if (idx < segEnd) {
            const float    v   = xval[idx];
            const unsigned rp1 = (unsigned)(xrow[idx] - base + 1);       // 1..16
            const unsigned g7  = ((unsigned)gidx[xcol[idx]]) << 7;       // g*128 < 2^22
            s_vm[tid] = make_uint2(__float_as_uint(v), (rp1 << 27) | g7);
            if (idx + CHUNK < segEnd) {         // global_prefetch_b8 next stage
                __builtin_prefetch(xval + idx + CHUNK, 0, 1);
                __builtin_prefetch(xcol + idx + CHUNK, 0, 1);
            }
        } else {                                // padding: val=0, rp1=0 never matches
            s_vm[tid] = make_uint2(0u, 0u);
        }
        __syncthreads();

        // Fixed trip count 64 (padded entries contribute 0) -> clean unrolling.
        #pragma unroll 8
        for (int k0 = kbeg; k0 < kbeg + CHUNK / 2; k0 += 4) {
            const int kA = k0 + ksel;           // even -> 16B-aligned pair
            // One ds_load_b128: {val0, meta0, val1, meta1}. The empty asm pins
            // all four dwords so the backend cannot scalarize the vector load
            // or predicate the val loads under v_cmpx (seen in round 2 asm).
            v4u q = *reinterpret_cast<const v4u*>(&s_vm[kA]);
            asm volatile("" : "+v"(q));

            const unsigned o0 = q.y & 0x07FFFFFFu;      // g0*128 (dword index)
            const unsigned r0 = q.y >> 27;
            const unsigned o1 = q.w & 0x07FFFFFFu;
            const unsigned r1 = q.w >> 27;

            v2f a, b;
            a.x = (r0 == m1) ? __uint_as_float(q.x) : 0.0f;   // A[m, kA]
            a.y = (r1 == m1) ? __uint_as_float(q.z) : 0.0f;   // A[m, kA+1]
            // 32-bit offsets -> saddr + scale_offset global_load_b32;
            // 64B-contiguous per 16-lane half
            b.x = embs[o0 + hcol];
            b.y = embs[o1 + hcol];

            acc = __builtin_amdgcn_wmma_f32_16x16x4_f32(
                /*neg_a=*/false, a, /*neg_b=*/false, b,
                /*c_mod=*/(short)0, acc, /*reuse_a=*/false, /*reuse_b=*/false);
        }
    }

    // Deterministic split-K combine: kg=1 parks its tile in LDS, kg=0 adds.
    // D layout: VGPR r -> M = r + 8*(lane>=16), N = lane&15
    const int mrow = (lane >> 4) << 3;
    float* tile = &s_red[ht * 256];
    if (kg == 1) {
        #pragma unroll
        for (int r = 0; r < 8; ++r)
            tile[(r + mrow) * 16 + (lane & 15)] = acc[r];
    }
    __syncthreads();
    if (kg == 0) {
        float* o = out + (size_t)(base + mrow) * NUM_HID_C + ht * 16 + (lane & 15);
        #pragma unroll
        for (int r = 0; r < 8; ++r)
            o[(size_t)r * NUM_HID_C] = acc[r] + tile[(r + mrow) * 16 + (lane & 15)];
    }
}

extern "C" void kernel_launch(void* const* d_in, const int* in_sizes, int n_in,
                              void* d_out, int out_size, void* d_ws, size_t ws_size,
                              hipStream_t stream) {
    const float* xval = (const float*)d_in[0];   // x_values  f32 [NNZ]
    const int*   xrow = (const int*)  d_in[1];   // x_rows    i32 [NNZ] (sorted)
    const int*   xcol = (const int*)  d_in[2];   // x_cols    i32 [NNZ]
    const int*   gidx = (const int*)  d_in[3];   // gene_idx  i32 [20000]
    const float* embs = (const float*)d_in[4];   // embs      f32 [30000,128]
    float*       outp = (float*)d_out;           // out       f32 [4096,128]
    const int nnz = in_sizes[0];

    dim3 grid(N_CELLS_C / CELLS_PER_BLOCK);      // 256 blocks of 16 cells
    omics_spmm_wmma<<<grid, THREADS, 0, stream>>>(xval, xrow, xcol, gidx, embs, outp, nnz);
}